// TopKConnectedLayer_69157563400534
// MI455X (gfx1250) — compile-verified
//
#include <hip/hip_runtime.h>
#include <stdint.h>

#define BB 8
#define NN 2000
#define EE 20000
#define SS (2*EE)
#define TPB 1024
#define EPT 20            // ceil(EE / TPB)
#define NEGF (-1e30f)

typedef uint32_t u32;
typedef uint64_t u64;
typedef u32 v4u __attribute__((ext_vector_type(4)));
typedef u32 v8u __attribute__((ext_vector_type(8)));

// Monotone float -> uint mapping (larger float => larger uint)
__device__ __forceinline__ u32 fkey(float x){
  u32 u = __float_as_uint(x);
  return (u & 0x80000000u) ? ~u : (u | 0x80000000u);
}
// 64-bit argmax key: value in high 32, (0x7FFFFFFF - idx) in low 32 so that
// exact-equal values break ties toward the LOWEST index (matches jnp.argmax).
__device__ __forceinline__ u64 packkey(float x, u32 idx){
  return ((u64)fkey(x) << 32) | (u64)(0x7FFFFFFFu - idx);
}
__device__ __forceinline__ u64 umax64(u64 a, u64 b){ return a > b ? a : b; }

// wave32 64-bit xor-shuffle (split into two 32-bit shuffles for portability)
__device__ __forceinline__ u64 shflxor64(u64 x, int m){
  u32 lo = __shfl_xor((u32)x, m, 32);
  u32 hi = __shfl_xor((u32)(x >> 32), m, 32);
  return ((u64)hi << 32) | lo;
}

#if defined(__gfx1250__)
// Tensor Data Mover: 1-D tile (nelem 32-bit words) global -> LDS.
// D# per CDNA5 ISA:
//  group0: [1:0]=count=1, [63:32]=lds_addr, [120:64]=global_addr, [127:126]=type=2
//  group1: data_size=2 (4B), tensor_dim0=nelem, tensor_dim1=1, tile_dim0=nelem,
//          tensor_dim0_stride=nelem, everything else 0 (no pad/iterate/multicast)
__device__ __forceinline__ void tdm_load_1d(const void* gsrc, u32 lds_byte_addr, u32 nelem){
  u64 ga = (u64)(uintptr_t)gsrc;
  v4u g0; v8u g1;
  g0[0] = 1u;                                                // count=1
  g0[1] = lds_byte_addr;                                     // lds_addr
  g0[2] = (u32)ga;                                           // global_addr[31:0]
  g0[3] = ((u32)(ga >> 32) & 0x01FFFFFFu) | 0x80000000u;     // addr[56:32] | type=2
  g1[0] = 0x00020000u;                                       // data_size=4B, mask=0
  g1[1] = (nelem & 0xFFFFu) << 16;                           // tensor_dim0[15:0]
  g1[2] = (nelem >> 16) | (1u << 16);                        // dim0[31:16] | tensor_dim1=1
  g1[3] = (nelem & 0xFFFFu) << 16;                           // tile_dim0
  g1[4] = 0u;                                                // tile_dim1/2 unused
  g1[5] = nelem;                                             // tensor_dim0_stride[31:0]
  g1[6] = 0u;
  g1[7] = 0u;
  asm volatile("tensor_load_to_lds %0, %1" :: "s"(g0), "s"(g1) : "memory");
}
#endif

// ---------------------------------------------------------------------------
// Kernel 1: one workgroup per graph. Stage logits/src/dst into LDS via TDM,
// then run the sequential connectivity-constrained top-k with per-thread
// register-resident edge state and a wave32-shuffle + ds_max_u64 argmax.
// Writes z (0/1 floats, EE per graph) to workspace.
// ---------------------------------------------------------------------------
extern "C" __global__ void __launch_bounds__(TPB)
topk_select_kernel(const float* __restrict__ logits,
                   const int*   __restrict__ ei,
                   const float* __restrict__ sample,
                   const int*   __restrict__ kptr,
                   float*       __restrict__ zws)
{
  extern __shared__ char smem[];
  float* gL = (float*)smem;                                   // EE floats
  int*   sL = (int*)(smem + (size_t)EE*4);                    // EE ints
  int*   dL = (int*)(smem + (size_t)EE*8);                    // EE ints
  unsigned long long* red = (unsigned long long*)(smem + (size_t)EE*12);
  u32* uvb = (u32*)(smem + (size_t)EE*12 + 8);

  const int b   = blockIdx.x;
  const int tid = threadIdx.x;

#if defined(__gfx1250__)
  if (tid < 32) {                       // wave 0 issues the three DMAs
    u32 base = (u32)(uintptr_t)smem;    // flat LDS addr low 32 bits == LDS offset
    tdm_load_1d(logits + (size_t)b*SS,                    base,         EE);
    tdm_load_1d(ei     + (size_t)b*SS,                    base + EE*4,  EE);
    tdm_load_1d(ei     + (size_t)BB*SS + (size_t)b*SS,    base + EE*8,  EE);
    __builtin_amdgcn_s_wait_tensorcnt(0);
  }
#else
  for (int e = tid; e < EE; e += TPB){
    gL[e] = logits[(size_t)b*SS + e];
    sL[e] = ei[(size_t)b*SS + e];
    dL[e] = ei[(size_t)BB*SS + (size_t)b*SS + e];
  }
#endif
  if (tid == 0) red[0] = 0ull;
  __syncthreads();

  // Per-thread register-resident edge state (strided assignment e = tid + j*TPB)
  float gR[EPT]; u32 sdR[EPT];
  #pragma unroll
  for (int j = 0; j < EPT; j++){
    int e = tid + j*TPB;
    if (e < EE){
      gR[j]  = gL[e] + sample[(size_t)b*EE + e];
      sdR[j] = ((u32)sL[e] & 0xFFFFu) | (((u32)dL[e] & 0xFFFFu) << 16);
    } else { gR[j] = NEGF; sdR[j] = 0xFFFFFFFFu; }   // 0xFFFF never matches a node id
  }

  const u32 negkey = fkey(NEGF);
  const int k = kptr[0];
  u32 act = 0, sel = 0;          // activated (eligible-ever) / selected bitmasks
  u32 u = 0xFFFFu, v = 0xFFFFu;  // newly marked nodes from previous selection
  u64 best = 0;                  // running max key over act & ~sel edges

  for (int i = 0; i < k; i++){
    u64 key;
    if (i == 0){
      // first pick: plain argmax over raw g (no connectivity constraint)
      key = 0;
      #pragma unroll
      for (int j = 0; j < EPT; j++){
        int e = tid + j*TPB;
        if (e < EE) key = umax64(key, packkey(gR[j], (u32)e));
      }
    } else {
      // incremental activation against the two newly marked nodes
      #pragma unroll
      for (int j = 0; j < EPT; j++){
        u32 bit = 1u << j;
        if (!((act | sel) & bit)){
          u32 sd = sdR[j];
          u32 s0 = sd & 0xFFFFu, d0 = sd >> 16;
          if (s0 == u || s0 == v || d0 == u || d0 == v){
            act  |= bit;
            best  = umax64(best, packkey(gR[j], (u32)(tid + j*TPB)));
          }
        }
      }
      key = best;
    }

    // wave32 reduce then one LDS u64 atomic-max per wave
    #pragma unroll
    for (int m = 16; m > 0; m >>= 1) key = umax64(key, shflxor64(key, m));
    if ((tid & 31) == 0) atomicMax(red, (unsigned long long)key);
    __syncthreads();

    u64 w = red[0];
    bool done = ((u32)(w >> 32)) <= negkey;   // no eligible edge left
    if (!done){
      u32 widx = 0x7FFFFFFFu - (u32)w;
      if ((int)(widx % TPB) == tid){          // owner of winning edge
        int slot = (int)(widx / TPB);
        uvb[0] = sdR[slot];                   // broadcast its endpoints
        sel   |= 1u << slot;
        best = 0;                             // recompute my running max
        #pragma unroll
        for (int j = 0; j < EPT; j++)
          if ((act & ~sel) & (1u << j))
            best = umax64(best, packkey(gR[j], (u32)(tid + j*TPB)));
      }
    }
    if (tid == 0) red[0] = 0ull;
    __syncthreads();
    if (done) break;                          // uniform across block
    u32 uv = uvb[0];
    u = uv & 0xFFFFu; v = uv >> 16;
  }

  // emit z from the selection bitmask (single writer per element, no pre-zero)
  #pragma unroll
  for (int j = 0; j < EPT; j++){
    int e = tid + j*TPB;
    if (e < EE) zws[(size_t)b*EE + e] = ((sel >> j) & 1u) ? 1.0f : 0.0f;
  }
}

// ---------------------------------------------------------------------------
// Kernel 2: to_undirected gather. rows==ei0 section, cols==ei1 section by
// construction; keys are unique, so out = v[lexsort] is computed via degree
// histogram + bucket scatter + per-node insertion sort, entirely in LDS.
// ---------------------------------------------------------------------------
extern "C" __global__ void __launch_bounds__(TPB)
undirect_gather_kernel(const int*   __restrict__ ei,
                       const float* __restrict__ zws,
                       float*       __restrict__ out)
{
  extern __shared__ char smem2[];
  int* deg   = (int*)smem2;
  int* start = (int*)(smem2 + (size_t)NN*4);
  int* cnt   = (int*)(smem2 + (size_t)NN*8);
  unsigned short* tmp = (unsigned short*)(smem2 + (size_t)NN*12);  // SS entries

  const int b   = blockIdx.x;
  const int tid = threadIdx.x;
  const int* r = ei + (size_t)b*SS;                      // rows
  const int* c = ei + (size_t)BB*SS + (size_t)b*SS;      // cols

  for (int x = tid; x < NN; x += TPB){ deg[x] = 0; cnt[x] = 0; }
  __syncthreads();
  for (int p = tid; p < SS; p += TPB) atomicAdd(&deg[r[p]], 1);
  __syncthreads();
  if (tid == 0){
    int run = 0;
    for (int x = 0; x < NN; x++){ start[x] = run; run += deg[x]; }
  }
  __syncthreads();
  for (int p = tid; p < SS; p += TPB){
    int x = r[p], y = c[p];
    int q = (p < EE) ? p : p - EE;                       // v = concat(z, z)
    int zb = (zws[(size_t)b*EE + q] > 0.5f) ? 1 : 0;
    int slot = start[x] + atomicAdd(&cnt[x], 1);
    tmp[slot] = (unsigned short)(((u32)y << 1) | (u32)zb);
  }
  __syncthreads();
  for (int x = tid; x < NN; x += TPB){                   // sort each bucket by col
    int s0 = start[x], d = deg[x];
    for (int a = 1; a < d; a++){
      unsigned short val = tmp[s0 + a];
      int t = a - 1;
      while (t >= 0 && tmp[s0 + t] > val){ tmp[s0 + t + 1] = tmp[s0 + t]; t--; }
      tmp[s0 + t + 1] = val;
    }
  }
  __syncthreads();
  for (int p = tid; p < SS; p += TPB)
    out[(size_t)b*SS + p] = (float)(tmp[p] & 1u);
}

// ---------------------------------------------------------------------------
extern "C" void kernel_launch(void* const* d_in, const int* in_sizes, int n_in,
                              void* d_out, int out_size, void* d_ws, size_t ws_size,
                              hipStream_t stream)
{
  (void)in_sizes; (void)n_in; (void)out_size; (void)ws_size;
  const float* logits = (const float*)d_in[0];
  const int*   ei     = (const int*)d_in[1];
  // d_in[2] = mask : unused (selection slice is structural)
  const float* sample = (const float*)d_in[3];
  const int*   kptr   = (const int*)d_in[4];
  float* out = (float*)d_out;
  float* zws = (float*)d_ws;                 // B*E floats

  const size_t lds1 = (size_t)EE*12 + 16;                 // ~234.4 KB (<320 KB WGP LDS)
  const size_t lds2 = (size_t)NN*12 + (size_t)SS*2;       // ~101.6 KB
  hipFuncSetAttribute((const void*)topk_select_kernel,
                      hipFuncAttributeMaxDynamicSharedMemorySize, (int)lds1);
  hipFuncSetAttribute((const void*)undirect_gather_kernel,
                      hipFuncAttributeMaxDynamicSharedMemorySize, (int)lds2);

  hipLaunchKernelGGL(topk_select_kernel, dim3(BB), dim3(TPB), lds1, stream,
                     logits, ei, sample, kptr, zws);
  hipLaunchKernelGGL(undirect_gather_kernel, dim3(BB), dim3(TPB), lds2, stream,
                     ei, zws, out);
}